// SVD_9990093931239
// MI455X (gfx1250) — compile-verified
//
#include <hip/hip_runtime.h>
#include <hip/hip_bf16.h>
#include <math.h>

// MI455X / gfx1250, wave32. One workgroup per batch element.
// Pipeline: async global->LDS stage (ASYNCcnt) | WMMA f32 16x16x4 Gram+moment
// accumulation | LDS cross-wave reduction | per-batch 3x3 Jacobi SVD | R,t.

typedef __attribute__((ext_vector_type(2))) float v2f;
typedef __attribute__((ext_vector_type(8))) float v8f;

#define THREADS 256
#define WAVES (THREADS / 32)

__global__ void __launch_bounds__(THREADS)
procrustes_gram_svd(const float* __restrict__ src,   // [B, N, 3]
                    const float* __restrict__ pose,  // [B, 3]
                    float* __restrict__ outR,        // [B, 3, 3]
                    float* __restrict__ outT,        // [B, 3]
                    int N)
{
    extern __shared__ float smem[];          // 3*N pts | WAVES*16 partial | 16 final
    float* pts = smem;
    float* red = smem + 3 * N;
    float* fin = red + WAVES * 16;

    const int b    = blockIdx.x;
    const int tid  = threadIdx.x;
    const int lane = tid & 31;
    const int wave = tid >> 5;

    // ---------------- Stage batch points into LDS ---------------------------
    // Direct global->LDS async copy (GVS mode: saddr base + 32-bit vgpr
    // offset). LDS destination offset = low 32 bits of the generic pointer
    // (LDS aperture: flat addr[31:0] == wave-relative LDS byte address).
    const float* gbase = src + (size_t)b * 3 * N;
    const unsigned ldsBase = (unsigned)(unsigned long long)(const void*)pts;
    const int nvec = (3 * N) >> 2;           // 1536 float4s, 6 per thread
    for (int i = tid; i < nvec; i += THREADS) {
        const unsigned voff = (unsigned)i * 16u;
        asm volatile("global_load_async_to_lds_b128 %0, %1, %2"
                     :
                     : "v"(ldsBase + voff), "v"(voff),
                       "s"((unsigned long long)gbase)
                     : "memory");
    }
#if __has_builtin(__builtin_amdgcn_s_wait_asynccnt)
    __builtin_amdgcn_s_wait_asynccnt(0);
#else
    asm volatile("s_wait_asynccnt 0x0" ::: "memory");
#endif
    __syncthreads();

    // ---------------- WMMA Gram + first-moment accumulation -----------------
    // A (16x4, MxK): VGPR v, lane L -> M = L%16, K = 2*(L/16)+v.  B (4x16)
    // mirrors it; the Gram X*X^T uses identical data in A and B, so one
    // per-lane fragment feeds both operands. Rows/cols 0..2 = channels,
    // row/col 3 = ones (accumulates sums S and the count in-instruction).
    const int  half = lane >> 4;             // K pair {0,1} vs {2,3}
    const int  ch   = lane & 15;             // matrix row/col index
    const bool isch = (ch < 3);
    const int  chc  = isch ? ch : 0;         // clamped address -> no exec mask
    const float fill = (ch == 3) ? 1.0f : 0.0f;

    v8f acc = {0.f, 0.f, 0.f, 0.f, 0.f, 0.f, 0.f, 0.f};

    const int ppw = N / WAVES;               // points per wave (256)
    int idx = (wave * ppw + 2 * half) * 3 + chc;
#pragma unroll 8
    for (int p = 0; p < ppw; p += 4) {
        // Unconditional LDS loads: scheduler can batch these ahead of the
        // WMMA chain instead of a per-WMMA exec-masked load + full wait.
        const float r0 = pts[idx];
        const float r1 = pts[idx + 3];
        idx += 12;
        v2f a;
        a.x = isch ? r0 : fill;
        a.y = isch ? r1 : fill;
        // D = A*B + C ; A == B (Gram). 8-arg form:
        // (neg_a, A, neg_b, B, c_mod, C, reuse_a, reuse_b)
        acc = __builtin_amdgcn_wmma_f32_16x16x4_f32(
                  false, a, false, a, (short)0, acc, false, false);
    }

    // C/D layout: VGPR r, lane L -> M = r + 8*(L/16), N = L%16.
    // The 4x4 block we need lives in VGPRs 0..3 of lanes 0..3.
    if (lane < 4) {
#pragma unroll
        for (int m = 0; m < 4; ++m)
            red[wave * 16 + m * 4 + lane] = acc[m];
    }
    __syncthreads();

    if (tid < 16) {
        float s = 0.0f;
#pragma unroll
        for (int w = 0; w < WAVES; ++w) s += red[w * 16 + tid];
        fin[tid] = s;
    }
    __syncthreads();

    // ---------------- Per-batch 3x3 solve (single lane) ---------------------
    if (tid == 0) {
        const float invN = 1.0f / (float)N;
        float P[3][3], S[3], mean[3];
#pragma unroll
        for (int i = 0; i < 3; ++i) {
#pragma unroll
            for (int j = 0; j < 3; ++j) P[i][j] = fin[i * 4 + j];
            S[i]    = fin[12 + i];       // row 3 of D = column sums
            mean[i] = S[i] * invN;
        }

        // Centered covariance H = P - S S^T / N  (symmetric PSD)
        float H[3][3];
#pragma unroll
        for (int i = 0; i < 3; ++i)
#pragma unroll
            for (int j = 0; j < 3; ++j)
                H[i][j] = P[i][j] - S[i] * S[j] * invN;

        // Cyclic Jacobi eigendecomposition: H = Q diag Q^T ; U = V = Q.
        float Q[3][3] = {{1.f, 0.f, 0.f}, {0.f, 1.f, 0.f}, {0.f, 0.f, 1.f}};
        const int PR[3] = {0, 0, 1};
        const int QR[3] = {1, 2, 2};
        for (int sweep = 0; sweep < 8; ++sweep) {
#pragma unroll
            for (int r = 0; r < 3; ++r) {
                const int p = PR[r], q = QR[r];
                const float apq = H[p][q];
                if (fabsf(apq) > 1e-12f * (fabsf(H[p][p]) + fabsf(H[q][q]) + 1e-30f)) {
                    const float tau = (H[q][q] - H[p][p]) / (2.0f * apq);
                    const float rt  = sqrtf(1.0f + tau * tau);
                    const float t   = (tau >= 0.0f) ? 1.0f / (tau + rt)
                                                    : 1.0f / (tau - rt);
                    const float c = 1.0f / sqrtf(1.0f + t * t);
                    const float s = t * c;
#pragma unroll
                    for (int k = 0; k < 3; ++k) {   // H = J^T H (rows p,q)
                        const float hp = H[p][k], hq = H[q][k];
                        H[p][k] = c * hp - s * hq;
                        H[q][k] = s * hp + c * hq;
                    }
#pragma unroll
                    for (int k = 0; k < 3; ++k) {   // H = H J (cols p,q)
                        const float hp = H[k][p], hq = H[k][q];
                        H[k][p] = c * hp - s * hq;
                        H[k][q] = s * hp + c * hq;
                    }
#pragma unroll
                    for (int k = 0; k < 3; ++k) {   // Q = Q J
                        const float qp = Q[k][p], qq = Q[k][q];
                        Q[k][p] = c * qp - s * qq;
                        Q[k][q] = s * qp + c * qq;
                    }
                }
            }
        }

        // R0 = V U^T = Q Q^T
        float R[3][3];
#pragma unroll
        for (int i = 0; i < 3; ++i)
#pragma unroll
            for (int j = 0; j < 3; ++j)
                R[i][j] = Q[i][0] * Q[j][0] + Q[i][1] * Q[j][1] + Q[i][2] * Q[j][2];

        const float det =
            R[0][0] * (R[1][1] * R[2][2] - R[1][2] * R[2][1]) -
            R[0][1] * (R[1][0] * R[2][2] - R[1][2] * R[2][0]) +
            R[0][2] * (R[1][0] * R[2][1] - R[1][1] * R[2][0]);

        if (det < 0.0f) {
            // Faithful to reference: Rr = (V @ reflect @ U^T) * reflect,
            // with the final '*' ELEMENTWISE against the dense diag matrix
            // (zeroes off-diagonals).
            const float refl[3] = {1.0f, 1.0f, -1.0f};
            float Rr[3][3];
#pragma unroll
            for (int i = 0; i < 3; ++i)
#pragma unroll
                for (int j = 0; j < 3; ++j) {
                    const float v = Q[i][0] * Q[j][0] + Q[i][1] * Q[j][1]
                                  - Q[i][2] * Q[j][2];
                    Rr[i][j] = (i == j) ? v * refl[i] : 0.0f;
                }
#pragma unroll
            for (int i = 0; i < 3; ++i)
#pragma unroll
                for (int j = 0; j < 3; ++j) R[i][j] = Rr[i][j];
        }

        // t = -R @ src_mean + tt_mean,  tt_mean = src_mean + pose
        const float pb[3] = {pose[b * 3 + 0], pose[b * 3 + 1], pose[b * 3 + 2]};
#pragma unroll
        for (int i = 0; i < 3; ++i) {
            const float rm = R[i][0] * mean[0] + R[i][1] * mean[1] + R[i][2] * mean[2];
            outT[b * 3 + i] = -rm + mean[i] + pb[i];
        }
#pragma unroll
        for (int i = 0; i < 3; ++i)
#pragma unroll
            for (int j = 0; j < 3; ++j)
                outR[b * 9 + i * 3 + j] = R[i][j];
    }
}

extern "C" void kernel_launch(void* const* d_in, const int* in_sizes, int n_in,
                              void* d_out, int out_size, void* d_ws, size_t ws_size,
                              hipStream_t stream)
{
    const float* src  = (const float*)d_in[0];   // source [B,N,3]
    // d_in[1] (template) is unused by the reference forward — never read it.
    const float* pose = (const float*)d_in[2];   // pose [B,3,1]

    const int B = in_sizes[2] / 3;               // 4096
    const int N = in_sizes[0] / (3 * B);         // 2048

    float* outR = (float*)d_out;                 // [B,3,3] flat
    float* outT = outR + (size_t)B * 9;          // [B,3] flat

    const size_t smemBytes = (size_t)(3 * N + WAVES * 16 + 16) * sizeof(float);
    procrustes_gram_svd<<<B, THREADS, smemBytes, stream>>>(src, pose, outR, outT, N);
}